// GraphLSTM_VAE_3599182594884
// MI455X (gfx1250) — compile-verified
//
#include <hip/hip_runtime.h>

// Problem constants (from reference)
constexpr int Bc = 64;    // batch
constexpr int Tc = 64;    // timesteps
constexpr int Nc = 32;    // nodes
constexpr int Dc = 32;    // feature dim
constexpr int Hc = 128;   // hidden
constexpr int Ec = 256;   // edges
constexpr int ROWS = Bc * Nc;        // 2048 GEMM rows
constexpr int GATES = 4 * Hc;        // 512 GEMM cols

typedef __attribute__((ext_vector_type(2))) float v2f;
typedef __attribute__((ext_vector_type(8))) float v8f;

// ---------------------------------------------------------------------------
// CDNA5 async global->LDS copy (ASYNCcnt-tracked), per-lane 16B transfer.
// ---------------------------------------------------------------------------
__device__ __forceinline__ void async_b128(unsigned lds_off, const void* g) {
  asm volatile("global_load_async_to_lds_b128 %0, %1, off"
               :: "v"(lds_off), "v"((unsigned long long)(uintptr_t)g)
               : "memory");
}
__device__ __forceinline__ void wait_async_le4() {
  asm volatile("s_wait_asynccnt 4" ::: "memory");
}
__device__ __forceinline__ void wait_async_0() {
  asm volatile("s_wait_asynccnt 0" ::: "memory");
}

__device__ __forceinline__ float sigmoidf_(float x) { return 1.f / (1.f + __expf(-x)); }

// ---------------------------------------------------------------------------
// Normalized adjacency: A = D^{-1/2} (A + I) D^{-1/2}, A[dst,src] += 1
// ---------------------------------------------------------------------------
__global__ __launch_bounds__(1024) void build_adj(const int* __restrict__ ei,
                                                  float* __restrict__ Aout) {
  __shared__ float sA[Nc * Nc];
  __shared__ float dinv[Nc];
  const int tid = threadIdx.x;  // 1024 threads == Nc*Nc
  sA[tid] = 0.f;
  __syncthreads();
  if (tid < Ec) {
    int src = ei[tid];       // edge_index[0, e]
    int dst = ei[Ec + tid];  // edge_index[1, e]
    atomicAdd(&sA[dst * Nc + src], 1.0f);
  }
  __syncthreads();
  if (tid < Nc) {
    sA[tid * Nc + tid] += 1.0f;
    float s = 0.f;
    for (int j = 0; j < Nc; ++j) s += sA[tid * Nc + j];
    dinv[tid] = rsqrtf(s);
  }
  __syncthreads();
  Aout[tid] = dinv[tid >> 5] * sA[tid] * dinv[tid & 31];
}

// ---------------------------------------------------------------------------
// x-part graph mix: out[t][b*32+n][c] = sum_j A[n,j] * src(b,t,j,c)
// src element (b,t,j,c) at src + b*bStride + t*tStride + j*32 + c  (D = 32)
// Encoder: one launch covers all T timesteps (off the critical path).
// Decoder: nT = 1 per step (mix of the fed-back output slice).
// ---------------------------------------------------------------------------
__global__ __launch_bounds__(256) void amix_x(const float* __restrict__ A,
                                              const float* __restrict__ src,
                                              long bStride, long tStride, int nT,
                                              float* __restrict__ out) {
  long idx = (long)blockIdx.x * 256 + threadIdx.x;
  long total = (long)nT * ROWS * Dc;
  if (idx >= total) return;
  int c = (int)(idx & 31);
  long rt = idx >> 5;        // t*2048 + r
  long r = rt & 2047;        // r = b*32+n
  long t = rt >> 11;
  int n = (int)(r & 31);
  long b = r >> 5;
  const float* arow = A + n * Nc;
  const float* p = src + b * bStride + t * tStride + c;
  float acc = 0.f;
  for (int j = 0; j < Nc; ++j) acc += arow[j] * p[(long)j * Dc];
  out[idx] = acc;
}

// ---------------------------------------------------------------------------
// Fused GCN-LSTM cell:
//   z = [X0 | X1] @ W + bias   (rows 2048, cols 512 packed [i,f,g,o])
//   c' = sig(z_f)*c + sig(z_i)*tanh(z_g);  h' = sig(z_o)*tanh(c')
//   Hmix = A @ h'  (graph mix fused in epilogue, via WMMA)
//
// A-operand streams from two segments: X0 [2048,F0] pitch p0 (x-part or
// mixed-h0), X1 [2048,128] pitch p1 (mixed h). Segment boundary is
// 16-chunk aligned (F0 in {32,128}).
//
// Grid (16, 4): blockIdx.x -> 128-row tile (= 4 whole batches, row = b*32+n);
// blockIdx.y -> 32-wide k-slab. Each block computes cols {q*H + ks*32..+32}
// for all 4 gates, exchanges through LDS, applies gates, then graph-mixes h'
// (all 32 nodes of its 4 batches are in-block) with V_WMMA_F32_16X16X4_F32.
//
// LDS (64KB, phase-overlaid):
//   compute phase : sA 2x[128x20f] @0, sB 2x[16x128f] @20480
//   gate exchange : zs [128][128] @0
//   h-mix phase   : hs [128][pitch 33] @0   (h' stashed in regs across barrier)
// ---------------------------------------------------------------------------
__global__ __launch_bounds__(256) void gemm_lstm(
    const float* __restrict__ X0, int p0, int F0,
    const float* __restrict__ X1, int p1,
    const float* __restrict__ W, const float* __restrict__ bias,
    float* __restrict__ Cst, float* __restrict__ Hst,
    float* __restrict__ Hmix, const float* __restrict__ Aadj, int K) {
  __shared__ __align__(16) char smem[65536];
  const unsigned ldsBase = (unsigned)(uintptr_t)smem;  // low 32 bits = LDS offset

  const int tid = threadIdx.x;
  const int wid = tid >> 5;
  const int lane = tid & 31;
  const int rowbase = blockIdx.x * 128;
  const int ksbase = blockIdx.y * 32;  // k-slab within H
  const int wm = wid & 1;              // 64-row half
  const int wn = wid >> 1;             // gate index q = 0..3 (32-col slice)
  const int lr = lane & 15;
  const int kp = (lane >> 4) << 1;     // K pair 0 or 2 (ISA fragment layout)
  const int mhi = (lane >> 4) * 8;

  // staging decomposition (same every chunk)
  const int r_a = tid >> 1, c_a = (tid & 1) * 8;
  const int kr_b = tid >> 4, c8_b = (tid & 15) * 8;
  const int q_b = c8_b >> 5, cc_b = c8_b & 31;

  v8f acc[4][2];
  for (int mi = 0; mi < 4; ++mi)
    for (int ni = 0; ni < 2; ++ni)
      for (int e = 0; e < 8; ++e) acc[mi][ni][e] = 0.f;

  auto stage = [&](int k0, int buf) {
    const float* ga = (k0 < F0)
        ? X0 + (long)(rowbase + r_a) * p0 + k0 + c_a
        : X1 + (long)(rowbase + r_a) * p1 + (k0 - F0) + c_a;
    unsigned la = ldsBase + (unsigned)buf * 10240u + (unsigned)(r_a * 20 + c_a) * 4u;
    async_b128(la, ga);
    async_b128(la + 16u, ga + 4);
    const float* gb = W + (long)(k0 + kr_b) * GATES + q_b * Hc + ksbase + cc_b;
    unsigned lb = ldsBase + 20480u + (unsigned)buf * 8192u +
                  (unsigned)(kr_b * 128 + c8_b) * 4u;
    async_b128(lb, gb);
    async_b128(lb + 16u, gb + 4);
  };

  const int nch = K >> 4;
  stage(0, 0);
  for (int ch = 0; ch < nch; ++ch) {
    const int cur = ch & 1;
    if (ch + 1 < nch) {
      stage((ch + 1) << 4, cur ^ 1);
      wait_async_le4();  // async loads complete in order: current chunk landed
    } else {
      wait_async_0();
    }
    __syncthreads();
    const float* sAc = (const float*)(smem + cur * 10240);
    const float* sBc = (const float*)(smem + 20480 + cur * 8192);
    for (int kk = 0; kk < 16; kk += 4) {
      v2f afrag[4], bfrag[2];
      for (int mi = 0; mi < 4; ++mi) {
        const float* ap = &sAc[(wm * 64 + mi * 16 + lr) * 20 + kk + kp];
        afrag[mi].x = ap[0];
        afrag[mi].y = ap[1];
      }
      for (int ni = 0; ni < 2; ++ni) {
        const float* bp = &sBc[(kk + kp) * 128 + wn * 32 + ni * 16 + lr];
        bfrag[ni].x = bp[0];
        bfrag[ni].y = bp[128];
      }
      for (int mi = 0; mi < 4; ++mi)
        for (int ni = 0; ni < 2; ++ni)
          acc[mi][ni] = __builtin_amdgcn_wmma_f32_16x16x4_f32(
              false, afrag[mi], false, bfrag[ni], (short)0, acc[mi][ni],
              false, false);
    }
    __syncthreads();
  }

  // ---- gate exchange through LDS ----
  float* zs = (float*)smem;  // [128][128]: col = gate*32 + c
  for (int mi = 0; mi < 4; ++mi)
    for (int ni = 0; ni < 2; ++ni) {
      int col = wn * 32 + ni * 16 + lr;
      for (int v = 0; v < 8; ++v) {
        int row = wm * 64 + mi * 16 + v + mhi;
        zs[row * 128 + col] = acc[mi][ni][v];
      }
    }
  __syncthreads();

  // ---- LSTM update: 16 (row,c) elements per thread; h' stashed in regs ----
  float hn[16];
  for (int e = 0; e < 16; ++e) {
    int idx = e * 256 + tid;  // 0..4095
    int r = idx >> 5;         // local row 0..127
    int c = idx & 31;
    int kc = ksbase + c;
    float zi = zs[r * 128 + 0 * 32 + c] + bias[0 * Hc + kc];
    float zf = zs[r * 128 + 1 * 32 + c] + bias[1 * Hc + kc];
    float zg = zs[r * 128 + 2 * 32 + c] + bias[2 * Hc + kc];
    float zo = zs[r * 128 + 3 * 32 + c] + bias[3 * Hc + kc];
    long sidx = (long)(rowbase + r) * Hc + kc;
    float ig = sigmoidf_(zi);
    float fg = sigmoidf_(zf);
    float gg = tanhf(zg);
    float og = sigmoidf_(zo);
    float cn = fg * Cst[sidx] + ig * gg;
    Cst[sidx] = cn;
    float hv = og * tanhf(cn);
    Hst[sidx] = hv;
    hn[e] = hv;
  }
  __syncthreads();  // all zs reads done -> LDS reusable for h exchange

  // ---- stash h' into LDS: hs[r][c], pitch 33 (conflict-free col reads) ----
  float* hs = (float*)smem;
  for (int e = 0; e < 16; ++e) {
    int idx = e * 256 + tid;
    hs[(idx >> 5) * 33 + (idx & 31)] = hn[e];
  }
  __syncthreads();

  // ---- graph mix via WMMA: Hmix[b,n,kc] = sum_j A[n,j] * h'[b,j,kc] ----
  // 16 output tiles: (bb 0..3) x (nt 0..1) x (ct 0..1); 2 per wave, K = 32.
  for (int i = 0; i < 2; ++i) {
    int tt = wid * 2 + i;
    int bb = tt >> 2, nt = (tt >> 1) & 1, ct = tt & 1;
    v8f macc;
    for (int e = 0; e < 8; ++e) macc[e] = 0.f;
    for (int kk = 0; kk < 32; kk += 4) {
      v2f a, b;
      const float* ap = Aadj + (nt * 16 + lr) * Nc + kk + kp;
      a.x = ap[0];
      a.y = ap[1];
      const float* bp = &hs[(bb * 32 + kk + kp) * 33 + ct * 16 + lr];
      b.x = bp[0];
      b.y = bp[33];
      macc = __builtin_amdgcn_wmma_f32_16x16x4_f32(false, a, false, b, (short)0,
                                                   macc, false, false);
    }
    for (int v = 0; v < 8; ++v) {
      int row = rowbase + bb * 32 + nt * 16 + v + mhi;
      int kc = ksbase + ct * 16 + lr;
      Hmix[(long)row * Hc + kc] = macc[v];
    }
  }
}

// ---------------------------------------------------------------------------
// Per-node linear: out[b,n,o] = sum_k x[b,n,k]*W[n,k,o] + bias[n,o]
// One block per node. rows = B = 64, K = H = 128, outF in {128, 32}.
// ---------------------------------------------------------------------------
__global__ __launch_bounds__(256) void pnl_wmma(const float* __restrict__ Xall, long xB,
                                                const float* __restrict__ Wall,
                                                const float* __restrict__ Ball,
                                                float* __restrict__ Out, long oB,
                                                int outF) {
  const int n = blockIdx.x;
  const float* Wn = Wall + (long)n * Hc * outF;
  const float* Bn = Ball + (long)n * outF;
  const float* Xn = Xall + (long)n * Hc;
  float* On = Out + (long)n * outF;
  const int tid = threadIdx.x;
  const int wid = tid >> 5;
  const int lane = tid & 31;
  const int lr = lane & 15;
  const int kp = (lane >> 4) << 1;
  const int tilesN = outF >> 4;
  const int total = 4 * tilesN;  // 4 M-tiles of 16 rows (B=64)

  for (int tt = wid; tt < total; tt += 8) {
    int mi = tt / tilesN;
    int ni = tt % tilesN;
    v8f acc;
    for (int e = 0; e < 8; ++e) acc[e] = 0.f;
    for (int k0 = 0; k0 < Hc; k0 += 4) {
      v2f a, b;
      const float* ap = Xn + (long)(mi * 16 + lr) * xB + k0 + kp;
      a.x = ap[0];
      a.y = ap[1];
      const float* bp = Wn + (long)(k0 + kp) * outF + ni * 16 + lr;
      b.x = bp[0];
      b.y = bp[outF];
      acc = __builtin_amdgcn_wmma_f32_16x16x4_f32(false, a, false, b, (short)0,
                                                  acc, false, false);
    }
    const int mhi = (lane >> 4) * 8;
    for (int v = 0; v < 8; ++v) {
      int brow = mi * 16 + v + mhi;
      int o = ni * 16 + lr;
      On[(long)brow * oB + o] = acc[v] + Bn[o];
    }
  }
}

// ---------------------------------------------------------------------------
__global__ __launch_bounds__(256) void reparam(const float* __restrict__ mu,
                                               const float* __restrict__ lv,
                                               const float* __restrict__ eps,
                                               float* __restrict__ zl) {
  long i = (long)blockIdx.x * 256 + threadIdx.x;  // exact B*N*H
  zl[i] = mu[i] + eps[i] * __expf(lv[i]);
}

__global__ __launch_bounds__(256) void zero_kernel(float* __restrict__ p, long nelem) {
  long i = (long)blockIdx.x * 256 + threadIdx.x;
  if (i < nelem) p[i] = 0.f;
}

// ---------------------------------------------------------------------------
extern "C" void kernel_launch(void* const* d_in, const int* in_sizes, int n_in,
                              void* d_out, int out_size, void* d_ws, size_t ws_size,
                              hipStream_t stream) {
  (void)in_sizes; (void)n_in; (void)out_size; (void)ws_size;
  const float* ts   = (const float*)d_in[0];
  const int*   ei   = (const int*)d_in[1];
  const float* Wen0 = (const float*)d_in[2];
  const float* ben0 = (const float*)d_in[3];
  const float* Wen1 = (const float*)d_in[4];
  const float* ben1 = (const float*)d_in[5];
  const float* Wde0 = (const float*)d_in[6];
  const float* bde0 = (const float*)d_in[7];
  const float* Wde1 = (const float*)d_in[8];
  const float* bde1 = (const float*)d_in[9];
  const float* Wmu  = (const float*)d_in[10];
  const float* bmu  = (const float*)d_in[11];
  const float* Wlv  = (const float*)d_in[12];
  const float* blv  = (const float*)d_in[13];
  const float* Who  = (const float*)d_in[14];
  const float* bho  = (const float*)d_in[15];
  const float* Wholv= (const float*)d_in[16];
  const float* bholv= (const float*)d_in[17];
  const float* eps  = (const float*)d_in[18];

  const long S = (long)Bc * Nc * Hc;        // 262144 (one [B,N,H] buffer)
  const long BTND = (long)Bc * Tc * Nc * Dc;

  // d_out regions (reference return order)
  float* outO  = (float*)d_out;             // [B,T,N,D]
  float* zlatO = outO + BTND;               // [B,N,H]
  float* muO   = zlatO + S;                 // [B,N,H]
  float* lvO   = muO + S;                   // [B,N,H]
  float* olvO  = lvO + S;                   // [B,T,N,D]

  // workspace
  float* wsf       = (float*)d_ws;
  float* Aw        = wsf;                            // 1024
  float* xpart_enc = Aw + 1024;                      // [T][2048][32]
  float* xpart_dec = xpart_enc + (long)Tc * ROWS * Dc;  // [2048][32]
  float* stZ       = xpart_dec + (long)ROWS * Dc;    // zeroed region start
  float* h0  = stZ;            // raw h/c state [B,N,H]
  float* c0  = h0 + S;
  float* h1  = c0 + S;
  float* c1  = h1 + S;
  float* h0mB1 = c1 + S;       // mixed-h ping-pong (parity 1 read first -> zeroed)
  float* h1mB1 = h0mB1 + S;
  float* h0mB0 = h1mB1 + S;
  float* h1mB0 = h0mB0 + S;
  float* h0m[2] = {h0mB0, h0mB1};
  float* h1m[2] = {h1mB0, h1mB1};

  const long NH = (long)Nc * Hc;
  const long TND = (long)Tc * Nc * Dc;   // 65536
  const long ND = (long)Nc * Dc;         // 1024
  const dim3 gGrid(ROWS / 128, Hc / 32); // 16 x 4

  build_adj<<<1, 1024, 0, stream>>>(ei, Aw);
  zero_kernel<<<(unsigned)((6 * S + 255) / 256), 256, 0, stream>>>(h0, 6 * S);
  // all encoder x-parts in one launch (off the recurrence critical path)
  amix_x<<<(unsigned)(((long)Tc * ROWS * Dc + 255) / 256), 256, 0, stream>>>(
      Aw, ts, TND, ND, Tc, xpart_enc);

  // ---------------- encoder: 2 launches per step ----------------
  for (int t = 0; t < Tc; ++t) {
    int pr = t & 1, rp = pr ^ 1;
    gemm_lstm<<<gGrid, 256, 0, stream>>>(
        xpart_enc + (long)t * ROWS * Dc, Dc, Dc, h0m[rp], Hc,
        Wen0, ben0, c0, h0, h0m[pr], Aw, Dc + Hc);
    gemm_lstm<<<gGrid, 256, 0, stream>>>(
        h0m[pr], Hc, Hc, h1m[rp], Hc,
        Wen1, ben1, c1, h1, h1m[pr], Aw, 2 * Hc);
  }

  // ---------------- latent ----------------
  pnl_wmma<<<Nc, 256, 0, stream>>>(h1, NH, Wmu, bmu, muO, NH, Hc);
  pnl_wmma<<<Nc, 256, 0, stream>>>(h1, NH, Wlv, blv, lvO, NH, Hc);
  reparam<<<(unsigned)(S / 256), 256, 0, stream>>>(muO, lvO, eps, zlatO);
  pnl_wmma<<<Nc, 256, 0, stream>>>(zlatO, NH, Who, bho,
                                   outO + (long)(Tc - 1) * ND, TND, Dc);
  pnl_wmma<<<Nc, 256, 0, stream>>>(zlatO, NH, Wholv, bholv,
                                   olvO + (long)(Tc - 1) * ND, TND, Dc);

  // ---------------- decoder (reverse time, feeds back own output) ----------
  for (int s = 0; s < Tc - 1; ++s) {
    int g = Tc + s;          // global step (parity continues from encoder)
    int pr = g & 1, rp = pr ^ 1;
    int tprev = Tc - 1 - s;  // slot written in previous iteration
    int tcur = Tc - 2 - s;
    amix_x<<<(unsigned)(((long)ROWS * Dc + 255) / 256), 256, 0, stream>>>(
        Aw, outO + (long)tprev * ND, TND, 0, 1, xpart_dec);
    gemm_lstm<<<gGrid, 256, 0, stream>>>(
        xpart_dec, Dc, Dc, h0m[rp], Hc,
        Wde0, bde0, c0, h0, h0m[pr], Aw, Dc + Hc);
    gemm_lstm<<<gGrid, 256, 0, stream>>>(
        h0m[pr], Hc, Hc, h1m[rp], Hc,
        Wde1, bde1, c1, h1, h1m[pr], Aw, 2 * Hc);
    pnl_wmma<<<Nc, 256, 0, stream>>>(h1, NH, Who, bho,
                                     outO + (long)tcur * ND, TND, Dc);
    pnl_wmma<<<Nc, 256, 0, stream>>>(h1, NH, Wholv, bholv,
                                     olvO + (long)tcur * ND, TND, Dc);
  }
}